// MotivationAttLayer_46943992545844
// MI455X (gfx1250) — compile-verified
//
#include <hip/hip_runtime.h>
#include <hip/hip_bf16.h>

// ---------------------------------------------------------------------------
// GAT-style attention layer for MI455X (gfx1250, wave32).
//   Phase 1: k = h_src@Wk, v = h_src@Wv, q = h_dst@Wq  via V_WMMA_F32_16X16X4_F32
//   Phase 2: per-edge logits + segment softmax + weighted scatter (L2-resident)
// ---------------------------------------------------------------------------

typedef __attribute__((ext_vector_type(2))) float v2f;
typedef __attribute__((ext_vector_type(8))) float v8f;

#define DDIM 128
#define NEG_SLOPE 0.01f

// Order-preserving encoding of float into uint so we can use native
// GLOBAL_ATOMIC_MAX_U32 for the segment max (works for negatives too).
__device__ __forceinline__ unsigned enc_ord(float x) {
    unsigned u = __float_as_uint(x);
    return (u & 0x80000000u) ? ~u : (u | 0x80000000u);
}
__device__ __forceinline__ float dec_ord(unsigned u) {
    return (u & 0x80000000u) ? __uint_as_float(u & 0x7fffffffu)
                             : __uint_as_float(~u);
}

// --------------------------- init -----------------------------------------
__global__ void init_kernel(float* __restrict__ out, int n_out,
                            unsigned* __restrict__ menc,
                            float* __restrict__ denom, int n_dst) {
    int i = blockIdx.x * blockDim.x + threadIdx.x;
    if (i < n_out) out[i] = 0.0f;
    if (i < n_dst) { menc[i] = 0u; denom[i] = 0.0f; }
}

// --------------------------- WMMA GEMM ------------------------------------
// O = A(M x 128) * W(128 x 128), row-major. One wave owns one 16x16 tile.
// DUAL=true computes two GEMMs sharing the A fragments (k and v from h_src).
//
// A-fragment (16x4 f32, 2 VGPRs): lanes 0-15 -> M=lane, K = k+{0,1};
//                                 lanes 16-31 -> M=lane-16, K = k+{2,3}.
// B-fragment (4x16 f32, 2 VGPRs): lanes 0-15 -> N=lane, rows K=k+{0,1};
//                                 lanes 16-31 -> N=lane-16, rows K=k+{2,3}.
// C/D (16x16 f32, 8 VGPRs): VGPR r, lanes 0-15 -> M=r, lanes 16-31 -> M=r+8.
template <bool DUAL>
__global__ void gemm_wmma_f32(const float* __restrict__ A,
                              const float* __restrict__ W1,
                              const float* __restrict__ W2,
                              float* __restrict__ O1,
                              float* __restrict__ O2,
                              int M) {
    const int lane = threadIdx.x & 31;
    const int wave = (blockIdx.x * blockDim.x + threadIdx.x) >> 5;
    const int mtiles = (M + 15) >> 4;
    const int ntiles = DDIM / 16;
    if (wave >= mtiles * ntiles) return;   // wave-uniform: EXEC stays all-1s

    const int mt = wave / ntiles;
    const int nt = wave % ntiles;
    const int l15 = lane & 15;
    const int hi  = lane >> 4;             // which half-wave (K pair select)

    int row = mt * 16 + l15;
    if (row >= M) row = M - 1;             // clamp loads; stores are masked
    const float* __restrict__ arow = A + (size_t)row * DDIM;
    const int n = nt * 16 + l15;

    v8f acc1 = {};
    v8f acc2 = {};

#pragma unroll
    for (int k = 0; k < DDIM; k += 4) {
        const int ka = k + 2 * hi;
        v2f a;                              // contiguous 8B load per lane
        a.x = arow[ka];
        a.y = arow[ka + 1];

        v2f b1;
        b1.x = W1[(size_t)ka * DDIM + n];
        b1.y = W1[(size_t)(ka + 1) * DDIM + n];
        acc1 = __builtin_amdgcn_wmma_f32_16x16x4_f32(
            false, a, false, b1, (short)0, acc1, false, false);

        if (DUAL) {
            v2f b2;
            b2.x = W2[(size_t)ka * DDIM + n];
            b2.y = W2[(size_t)(ka + 1) * DDIM + n];
            acc2 = __builtin_amdgcn_wmma_f32_16x16x4_f32(
                false, a, false, b2, (short)0, acc2, false, false);
        }
    }

#pragma unroll
    for (int r = 0; r < 8; ++r) {
        int orow = mt * 16 + r + 8 * hi;
        if (orow < M) {
            O1[(size_t)orow * DDIM + n] = acc1[r];
            if (DUAL) O2[(size_t)orow * DDIM + n] = acc2[r];
        }
    }
}

// --------------------------- edge logits + segment max --------------------
// One wave per edge: 32 lanes x float4 = 128 floats, coalesced 512B rows.
__global__ void edge_logits_kernel(const float* __restrict__ kmat,
                                   const float* __restrict__ qmat,
                                   const int* __restrict__ src,
                                   const int* __restrict__ dst,
                                   float* __restrict__ e,
                                   unsigned* __restrict__ menc, int E) {
    const int lane = threadIdx.x & 31;
    const int edge = (blockIdx.x * blockDim.x + threadIdx.x) >> 5;
    if (edge >= E) return;

    const int s = src[edge];
    const int d = dst[edge];
    const float4 kk = *(const float4*)(kmat + (size_t)s * DDIM + lane * 4);
    const float4 qq = *(const float4*)(qmat + (size_t)d * DDIM + lane * 4);

    float sum = kk.x * qq.x + kk.y * qq.y + kk.z * qq.z + kk.w * qq.w;
    sum += __shfl_xor(sum, 16, 32);
    sum += __shfl_xor(sum, 8, 32);
    sum += __shfl_xor(sum, 4, 32);
    sum += __shfl_xor(sum, 2, 32);
    sum += __shfl_xor(sum, 1, 32);

    if (lane == 0) {
        float val = sum > 0.0f ? sum : NEG_SLOPE * sum;  // leaky_relu
        e[edge] = val;
        atomicMax(menc + d, enc_ord(val));               // native u32 atomic max
    }
}

// --------------------------- exp + segment sum ----------------------------
__global__ void edge_exp_kernel(float* __restrict__ e,
                                const int* __restrict__ dst,
                                const unsigned* __restrict__ menc,
                                float* __restrict__ denom, int E) {
    int i = blockIdx.x * blockDim.x + threadIdx.x;
    if (i >= E) return;
    const int d = dst[i];
    const float ex = __expf(e[i] - dec_ord(menc[d]));
    e[i] = ex;
    atomicAdd(denom + d, ex);                            // GLOBAL_ATOMIC_ADD_F32
}

// --------------------------- weighted scatter -----------------------------
// One wave per edge; each lane atomically accumulates 4 floats of the row.
__global__ void edge_scatter_kernel(const float* __restrict__ e,
                                    const float* __restrict__ denom,
                                    const float* __restrict__ vmat,
                                    const int* __restrict__ src,
                                    const int* __restrict__ dst,
                                    float* __restrict__ out, int E) {
    const int lane = threadIdx.x & 31;
    const int edge = (blockIdx.x * blockDim.x + threadIdx.x) >> 5;
    if (edge >= E) return;

    const int s = src[edge];
    const int d = dst[edge];
    const float alpha = e[edge] / denom[d];
    const float4 vv = *(const float4*)(vmat + (size_t)s * DDIM + lane * 4);
    float* o = out + (size_t)d * DDIM + lane * 4;
    atomicAdd(o + 0, alpha * vv.x);
    atomicAdd(o + 1, alpha * vv.y);
    atomicAdd(o + 2, alpha * vv.z);
    atomicAdd(o + 3, alpha * vv.w);
}

// --------------------------- host launch ----------------------------------
extern "C" void kernel_launch(void* const* d_in, const int* in_sizes, int n_in,
                              void* d_out, int out_size, void* d_ws, size_t ws_size,
                              hipStream_t stream) {
    const float* h_src = (const float*)d_in[0];
    const float* h_dst = (const float*)d_in[1];
    const float* Wk    = (const float*)d_in[2];
    const float* Wq    = (const float*)d_in[3];
    const float* Wv    = (const float*)d_in[4];
    const int* src_idx = (const int*)d_in[5];
    const int* dst_idx = (const int*)d_in[6];
    float* out = (float*)d_out;

    const int n_src = in_sizes[0] / DDIM;
    const int n_dst = in_sizes[1] / DDIM;
    const int E     = in_sizes[5];

    // Carve workspace (k, v, q, e, m_enc, denom) ~= 80 MB
    char* base = (char*)d_ws;
    size_t off = 0;
    auto carve = [&](size_t bytes) -> void* {
        void* p = base + off;
        off = (off + bytes + 255) & ~(size_t)255;
        return p;
    };
    float*    kbuf  = (float*)carve((size_t)n_src * DDIM * sizeof(float));
    float*    vbuf  = (float*)carve((size_t)n_src * DDIM * sizeof(float));
    float*    qbuf  = (float*)carve((size_t)n_dst * DDIM * sizeof(float));
    float*    ebuf  = (float*)carve((size_t)E * sizeof(float));
    unsigned* menc  = (unsigned*)carve((size_t)n_dst * sizeof(unsigned));
    float*    denom = (float*)carve((size_t)n_dst * sizeof(float));

    // 1) zero out/m/denom (harness poisons buffers; must re-init every call)
    {
        int n = out_size > n_dst ? out_size : n_dst;
        init_kernel<<<(n + 255) / 256, 256, 0, stream>>>(out, out_size, menc, denom, n_dst);
    }
    // 2) k,v = h_src @ {Wk, Wv} (dual-GEMM, shared A fragments)
    {
        int tiles = ((n_src + 15) / 16) * (DDIM / 16);
        int blocks = (tiles * 32 + 255) / 256;
        gemm_wmma_f32<true><<<blocks, 256, 0, stream>>>(h_src, Wk, Wv, kbuf, vbuf, n_src);
    }
    // 3) q = h_dst @ Wq
    {
        int tiles = ((n_dst + 15) / 16) * (DDIM / 16);
        int blocks = (tiles * 32 + 255) / 256;
        gemm_wmma_f32<false><<<blocks, 256, 0, stream>>>(h_dst, Wq, nullptr, qbuf, nullptr, n_dst);
    }
    // 4) per-edge logits + leaky_relu + segment max (one wave per edge)
    {
        int blocks = ((size_t)E * 32 + 255) / 256;
        edge_logits_kernel<<<blocks, 256, 0, stream>>>(kbuf, qbuf, src_idx, dst_idx, ebuf, menc, E);
    }
    // 5) exp(e - m[dst]) and segment sum of denominators
    {
        int blocks = (E + 255) / 256;
        edge_exp_kernel<<<blocks, 256, 0, stream>>>(ebuf, dst_idx, menc, denom, E);
    }
    // 6) out[dst] += (ex/denom[dst]) * v[src]  (one wave per edge)
    {
        int blocks = ((size_t)E * 32 + 255) / 256;
        edge_scatter_kernel<<<blocks, 256, 0, stream>>>(ebuf, denom, vbuf, src_idx, dst_idx, out, E);
    }
}